// MessagePassing_63685775065334
// MI455X (gfx1250) — compile-verified
//
#include <hip/hip_runtime.h>

#define IN_CH 64
#define EDGES_PER_WAVE 128

typedef float v2f __attribute__((ext_vector_type(2)));
typedef float v8f __attribute__((ext_vector_type(8)));

// ---------------------------------------------------------------------------
// Kernel 0: zero the output accumulator (needed every call: graph replays).
// ---------------------------------------------------------------------------
__global__ void zero_out_kernel(float4* __restrict__ out, int n4) {
  int i = blockIdx.x * blockDim.x + threadIdx.x;
  int stride = gridDim.x * blockDim.x;
  float4 z = make_float4(0.f, 0.f, 0.f, 0.f);
  for (; i < n4; i += stride) out[i] = z;
}

// ---------------------------------------------------------------------------
// Kernel 1: per-cell attention partials via WMMA.
//   y1[n] = dot(X[n], W[0:64]),  y2[n] = dot(X[n], W[64:128])
// One wave computes 16 rows. A = 16x4 f32 tile of X, B = 4x16 with
// column 0 = w1 chunk, column 1 = w2 chunk, other columns zero.
// 16 chained V_WMMA_F32_16X16X4_F32 accumulate K=64.
// ---------------------------------------------------------------------------
__global__ void gemv_attn_wmma(const float* __restrict__ X,
                               const float* __restrict__ W,
                               float* __restrict__ y1,
                               float* __restrict__ y2,
                               int n_cells) {
  const int lane = threadIdx.x & 31;
  const int wave = blockIdx.x * (blockDim.x >> 5) + (threadIdx.x >> 5);
  const int row_base = wave * 16;
  if (row_base >= n_cells) return;          // wave-uniform exit: EXEC stays full

  const int m  = lane & 15;                 // row within tile / column of B
  const int kh = (lane >> 4) << 1;          // K sub-offset: 0 or 2
  int row = row_base + m;
  if (row >= n_cells) row = n_cells - 1;    // clamp loads, keep EXEC all-ones
  const float* rowp = X + (size_t)row * IN_CH;

  // branch-free weight source for this lane's B column
  const float  bmask = (m <= 1) ? 1.0f : 0.0f;
  const float* wp    = W + ((m == 1) ? IN_CH : 0) + kh;

  v8f c = {};
#pragma unroll
  for (int k0 = 0; k0 < IN_CH; k0 += 4) {
    v2f a, b;
    a.x = rowp[k0 + kh];
    a.y = rowp[k0 + kh + 1];
    b.x = bmask * wp[k0];
    b.y = bmask * wp[k0 + 1];
    c = __builtin_amdgcn_wmma_f32_16x16x4_f32(false, a, false, b,
                                              (short)0, c, false, false);
  }

  // Column 0 of D = y1 (lanes 0 & 16), column 1 = y2 (lanes 1 & 17).
  if (m <= 1) {
    float* dst = (m == 0) ? y1 : y2;
    const int mb = row_base + ((lane >> 4) << 3);
#pragma unroll
    for (int r = 0; r < 8; ++r) {
      int rr = mb + r;
      if (rr < n_cells) dst[rr] = c[r];
    }
  }
}

// ---------------------------------------------------------------------------
// Kernel 2: per-edge attention + scatter-sum.
// One wave processes EDGES_PER_WAVE contiguous (target-sorted) edges.
// Lane l owns channels 2l, 2l+1. Metadata loaded 32 edges at a time
// (coalesced); per-edge broadcast uses v_readlane with constant lane index,
// so src/tgt/val land in SGPRs: scalar run-boundary branch, SADDR-form
// gathers, zero DS traffic. Register accumulation over runs of equal
// target; atomic f32 flush at run/chunk boundaries.
// ---------------------------------------------------------------------------
__global__ void edge_scatter_kernel(const float* __restrict__ X,
                                    const float* __restrict__ y1,
                                    const float* __restrict__ y2,
                                    const float* __restrict__ nbhd,
                                    const int* __restrict__ tgt_idx,
                                    const int* __restrict__ src_idx,
                                    float* __restrict__ out,
                                    int n_edges) {
  const int lane = threadIdx.x & 31;
  const int wave = blockIdx.x * (blockDim.x >> 5) + (threadIdx.x >> 5);
  const long base = (long)wave * EDGES_PER_WAVE;
  if (base >= n_edges) return;

  const int ch = lane << 1;  // first of this lane's 2 channels

  float accx = 0.f, accy = 0.f;
  int cur_tgt = -1;

  for (int it = 0; it < EDGES_PER_WAVE; it += 32) {
    long e = base + it + lane;
    const bool valid = (e < (long)n_edges);
    long ec = valid ? e : (long)n_edges - 1;

    int   s  = src_idx[ec];
    int   t  = tgt_idx[ec];
    float nv = nbhd[ec];

    // attention logit and ELU (alpha = 1)
    float z   = y1[s] + y2[t];
    float att = (z > 0.f) ? z : expm1f(z);
    float val = valid ? (nv * att) : 0.f;
    int   vbits = __float_as_int(val);

#pragma unroll
    for (int j = 0; j < 32; ++j) {
      // constant-lane readlane -> SGPR broadcast
      int   sj = __builtin_amdgcn_readlane(s, j);
      int   tj = __builtin_amdgcn_readlane(t, j);
      float v  = __int_as_float(__builtin_amdgcn_readlane(vbits, j));

      const float* xp = X + (size_t)sj * IN_CH;   // uniform base (SADDR)
      const float2 xs = *(const float2*)(xp + ch);

      if (tj != cur_tgt) {   // scalar compare + branch (uniform values)
        if (cur_tgt >= 0) {
          float* dst = out + (size_t)cur_tgt * IN_CH + ch;
          atomicAdd(dst + 0, accx);
          atomicAdd(dst + 1, accy);
        }
        cur_tgt = tj;
        accx = 0.f;
        accy = 0.f;
      }
      accx = fmaf(v, xs.x, accx);
      accy = fmaf(v, xs.y, accy);
    }
  }
  if (cur_tgt >= 0) {
    float* dst = out + (size_t)cur_tgt * IN_CH + ch;
    atomicAdd(dst + 0, accx);
    atomicAdd(dst + 1, accy);
  }
}

// ---------------------------------------------------------------------------
extern "C" void kernel_launch(void* const* d_in, const int* in_sizes, int n_in,
                              void* d_out, int out_size, void* d_ws, size_t ws_size,
                              hipStream_t stream) {
  const float* X    = (const float*)d_in[0];  // [n_cells, 64]
  const float* W    = (const float*)d_in[1];  // [128]
  const float* nbhd = (const float*)d_in[2];  // [n_edges]
  const int*   tgt  = (const int*)d_in[3];    // [n_edges] sorted
  const int*   src  = (const int*)d_in[4];    // [n_edges]
  float*       out  = (float*)d_out;          // [n_cells, 64]

  const int n_cells = in_sizes[0] / IN_CH;
  const int n_edges = in_sizes[2];

  float* y1 = (float*)d_ws;        // [n_cells]
  float* y2 = y1 + n_cells;        // [n_cells]

  // 0) zero output accumulator
  {
    int n4 = out_size / 4;
    int threads = 256;
    int blocks = (n4 + threads - 1) / threads;
    if (blocks > 6554) blocks = 6554;
    zero_out_kernel<<<blocks, threads, 0, stream>>>((float4*)out, n4);
  }

  // 1) per-cell attention partials (WMMA)
  {
    int waves = (n_cells + 15) / 16;
    int threads = 128;                        // 4 waves / block
    int blocks = (waves + 3) / 4;
    gemv_attn_wmma<<<blocks, threads, 0, stream>>>(X, W, y1, y2, n_cells);
  }

  // 2) edge attention + scatter
  {
    int waves = (n_edges + EDGES_PER_WAVE - 1) / EDGES_PER_WAVE;
    int threads = 256;                        // 8 waves / block
    int blocks = (waves + 7) / 8;
    edge_scatter_kernel<<<blocks, threads, 0, stream>>>(
        X, y1, y2, nbhd, tgt, src, out, n_edges);
  }
}